// ConvFlow_472446403276
// MI455X (gfx1250) — compile-verified
//
#include <hip/hip_runtime.h>
#include <math.h>

#define B_    16
#define CIN   4
#define HALFC 2
#define FILT  256
#define T_    8192
#define NLAY  3
#define KSZ   3
#define NBINS 10
#define TAILV 5.0f
#define MINWV 0.001f
#define TTILE 128
#define NTILES (T_/TTILE)   // 64
#define PROJ  58
#define TP    (TTILE + 1)   // padded column count per row-pair

typedef __attribute__((ext_vector_type(2))) float v2f;
typedef __attribute__((ext_vector_type(8))) float v8f;

// Row-pair interleaved LDS layout: (c,t) -> ((c>>1)*TP + t)*2 + (c&1).
// Rows 2i and 2i+1 at the same column are adjacent & 8B aligned, so a WMMA
// B-fragment {A[k][n], A[k+1][n]} (k always even) is ONE ds_load_b64.
__device__ __forceinline__ int sidx(int c, int t) {
  return (((c >> 1) * TP) + t) * 2 + (c & 1);
}

__device__ __forceinline__ v8f v8zero() {
  v8f v;
#pragma unroll
  for (int i = 0; i < 8; ++i) v[i] = 0.0f;
  return v;
}

__device__ __forceinline__ float gelu_exact(float v) {
  return 0.5f * v * (1.0f + erff(v * 0.70710678118654752f));
}

__device__ __forceinline__ float softplusf(float v) {
  return (v > 20.0f) ? v : log1pf(expf(v));
}

// --------------------------------------------------------------------------
// Kernel 1: h = pre_w @ x0 + pre_b
// --------------------------------------------------------------------------
__global__ void pre_kernel(const float* __restrict__ x,
                           const float* __restrict__ pw,
                           const float* __restrict__ pb,
                           float* __restrict__ h) {
  size_t i = (size_t)blockIdx.x * blockDim.x + threadIdx.x;
  const size_t total = (size_t)B_ * FILT * T_;
  if (i >= total) return;
  int t = (int)(i % T_);
  int o = (int)((i / T_) % FILT);
  int b = (int)(i / ((size_t)FILT * T_));
  float x0 = x[((size_t)b * CIN + 0) * T_ + t];
  float x1 = x[((size_t)b * CIN + 1) * T_ + t];
  h[i] = pw[o * 2 + 0] * x0 + pw[o * 2 + 1] * x1 + pb[o];
}

// --------------------------------------------------------------------------
// Kernel 2 (x3): dwconv -> LN -> GELU -> 256x256 WMMA-f32 matmul -> +b1
//               -> LN -> GELU -> residual.  One (batch, 128-col) tile / WG.
// --------------------------------------------------------------------------
__global__ __launch_bounds__(256, 1)
void layer_kernel(const float* __restrict__ hin, float* __restrict__ hout,
                  const float* __restrict__ xmask,
                  const float* __restrict__ sep_w, const float* __restrict__ sep_b,
                  const float* __restrict__ w1, const float* __restrict__ b1,
                  const float* __restrict__ g1, const float* __restrict__ be1,
                  const float* __restrict__ g2, const float* __restrict__ be2,
                  int l, int dil) {
  __shared__ float sA[128 * TP * 2];      // 256 x (128+pad) activation/z tile
  __shared__ float sS1[2][TTILE];
  __shared__ float sS2[2][TTILE];
  __shared__ float sMean[TTILE];
  __shared__ float sRstd[TTILE];

  const int tid  = threadIdx.x;
  const int b    = blockIdx.x / NTILES;
  const int tile = blockIdx.x % NTILES;
  const int t0   = tile * TTILE;

  // ---- Phase A: depthwise conv (h * mask), zero padding at T boundaries ----
  for (int idx = tid; idx < FILT * TTILE; idx += 256) {
    int c = idx >> 7;
    int t = idx & (TTILE - 1);
    int tg = t0 + t;
    const float* hrow = hin + ((size_t)b * FILT + c) * T_;
    const float* mrow = xmask + (size_t)b * T_;
    float w0 = sep_w[((l * FILT) + c) * KSZ + 0];
    float wc = sep_w[((l * FILT) + c) * KSZ + 1];
    float w2 = sep_w[((l * FILT) + c) * KSZ + 2];
    float acc = sep_b[l * FILT + c];
    int tm = tg - dil, tp = tg + dil;
    if (tm >= 0) acc += w0 * hrow[tm] * mrow[tm];
    acc += wc * hrow[tg] * mrow[tg];
    if (tp < T_) acc += w2 * hrow[tp] * mrow[tp];
    sA[sidx(c, t)] = acc;
  }
  __syncthreads();

  // ---- LN1 stats (channel-dim, per column) ----
  {
    int col = tid & (TTILE - 1);
    int hf  = tid >> 7;
    float s1 = 0.f, s2 = 0.f;
    for (int c = hf * 128; c < hf * 128 + 128; ++c) {
      float v = sA[sidx(c, col)]; s1 += v; s2 += v * v;
    }
    sS1[hf][col] = s1; sS2[hf][col] = s2;
  }
  __syncthreads();
  if (tid < TTILE) {
    float s1 = sS1[0][tid] + sS1[1][tid];
    float s2 = sS2[0][tid] + sS2[1][tid];
    float m  = s1 * (1.0f / FILT);
    float va = s2 * (1.0f / FILT) - m * m;
    sMean[tid] = m;
    sRstd[tid] = rsqrtf(fmaxf(va, 0.0f) + 1e-5f);
  }
  __syncthreads();
  for (int idx = tid; idx < FILT * TTILE; idx += 256) {
    int c = idx >> 7; int t = idx & (TTILE - 1);
    int si = sidx(c, t);
    float v = (sA[si] - sMean[t]) * sRstd[t] * g1[l * FILT + c] + be1[l * FILT + c];
    sA[si] = gelu_exact(v);
  }
  __syncthreads();

  // ---- Phase B: z = w1 @ a  via V_WMMA_F32_16X16X4_F32 ----
  const int wv   = tid >> 5;
  const int lane = tid & 31;
  const int m    = lane & 15;
  const int kk   = (lane >> 4) << 1;   // 0 or 2 (K pair select)
  const int mh   = (lane >> 4) << 3;   // D-tile row half

  v8f acc[2][8];
#pragma unroll
  for (int i = 0; i < 2; ++i)
#pragma unroll
    for (int j = 0; j < 8; ++j) acc[i][j] = v8zero();

#pragma unroll
  for (int obi = 0; obi < 2; ++obi) {
    int ob = wv * 2 + obi;
    const float* arow = w1 + ((size_t)(l * FILT) + ob * 16 + m) * FILT + kk;
    for (int kb = 0; kb < FILT / 4; ++kb) {
      int k0 = kb * 4;
      v2f afrag = *(const v2f*)(arow + k0);            // global_load_b64
      int rp = (k0 + kk) >> 1;                         // row pair (row even)
#pragma unroll
      for (int tb = 0; tb < 8; ++tb) {
        int col = tb * 16 + m;
        v2f bfrag = *(const v2f*)&sA[(rp * TP + col) * 2];  // ds_load_b64
        acc[obi][tb] = __builtin_amdgcn_wmma_f32_16x16x4_f32(
            false, afrag, false, bfrag, (short)0, acc[obi][tb], false, false);
      }
    }
  }
  __syncthreads();   // everyone done reading sA as activations

  // ---- write z + b1 back into sA ----
#pragma unroll
  for (int obi = 0; obi < 2; ++obi) {
    int ob = wv * 2 + obi;
#pragma unroll
    for (int tb = 0; tb < 8; ++tb) {
      int col = tb * 16 + m;
#pragma unroll
      for (int r = 0; r < 8; ++r) {
        int o = ob * 16 + mh + r;
        sA[sidx(o, col)] = acc[obi][tb][r] + b1[l * FILT + o];
      }
    }
  }
  __syncthreads();

  // ---- LN2 stats ----
  {
    int col = tid & (TTILE - 1);
    int hf  = tid >> 7;
    float s1 = 0.f, s2 = 0.f;
    for (int c = hf * 128; c < hf * 128 + 128; ++c) {
      float v = sA[sidx(c, col)]; s1 += v; s2 += v * v;
    }
    sS1[hf][col] = s1; sS2[hf][col] = s2;
  }
  __syncthreads();
  if (tid < TTILE) {
    float s1 = sS1[0][tid] + sS1[1][tid];
    float s2 = sS2[0][tid] + sS2[1][tid];
    float m  = s1 * (1.0f / FILT);
    float va = s2 * (1.0f / FILT) - m * m;
    sMean[tid] = m;
    sRstd[tid] = rsqrtf(fmaxf(va, 0.0f) + 1e-5f);
  }
  __syncthreads();

  // ---- epilogue: GELU(LN2(z)) + residual ----
  for (int idx = tid; idx < FILT * TTILE; idx += 256) {
    int c = idx >> 7; int t = idx & (TTILE - 1);
    float v = (sA[sidx(c, t)] - sMean[t]) * sRstd[t] * g2[l * FILT + c] + be2[l * FILT + c];
    v = gelu_exact(v);
    size_t gi = ((size_t)b * FILT + c) * T_ + t0 + t;
    hout[gi] = hin[gi] + v;
  }
}

// --------------------------------------------------------------------------
// Kernel 3: p = proj_w @ (h*mask) + proj_b (WMMA, o padded 58->64), then the
//           rational-quadratic spline on x1, xo output, logdet partials.
// --------------------------------------------------------------------------
__global__ __launch_bounds__(256, 1)
void proj_spline_kernel(const float* __restrict__ h, const float* __restrict__ x,
                        const float* __restrict__ xmask,
                        const float* __restrict__ pw, const float* __restrict__ pb,
                        float* __restrict__ out, float* __restrict__ part) {
  __shared__ float sA[128 * TP * 2];
  __shared__ float sRed[256];

  const int tid  = threadIdx.x;
  const int b    = blockIdx.x / NTILES;
  const int tile = blockIdx.x % NTILES;
  const int t0   = tile * TTILE;

  for (int idx = tid; idx < FILT * TTILE; idx += 256) {
    int c = idx >> 7; int t = idx & (TTILE - 1); int tg = t0 + t;
    sA[sidx(c, t)] = h[((size_t)b * FILT + c) * T_ + tg] * xmask[(size_t)b * T_ + tg];
  }
  __syncthreads();

  const int wv   = tid >> 5;
  const int lane = tid & 31;
  const int m    = lane & 15;
  const int kk   = (lane >> 4) << 1;
  const int mh   = (lane >> 4) << 3;
  const int tb   = wv;                 // each wave owns one 16-col block

  v8f acc[4];
#pragma unroll
  for (int j = 0; j < 4; ++j) acc[j] = v8zero();

  for (int kb = 0; kb < FILT / 4; ++kb) {
    int k0  = kb * 4;
    int rp  = (k0 + kk) >> 1;
    int col = tb * 16 + m;
    v2f bfrag = *(const v2f*)&sA[(rp * TP + col) * 2];   // ds_load_b64
#pragma unroll
    for (int obi = 0; obi < 4; ++obi) {
      int o = obi * 16 + m;
      v2f afrag;
      afrag.x = (o < PROJ) ? pw[(size_t)o * FILT + k0 + kk]     : 0.0f;
      afrag.y = (o < PROJ) ? pw[(size_t)o * FILT + k0 + kk + 1] : 0.0f;
      acc[obi] = __builtin_amdgcn_wmma_f32_16x16x4_f32(
          false, afrag, false, bfrag, (short)0, acc[obi], false, false);
    }
  }
  __syncthreads();   // done reading h tile

  { // write p (rows 0..63) into sA, with bias and mask
    int col = tb * 16 + m;
    float maskc = xmask[(size_t)b * T_ + t0 + col];
#pragma unroll
    for (int obi = 0; obi < 4; ++obi) {
#pragma unroll
      for (int r = 0; r < 8; ++r) {
        int o = obi * 16 + mh + r;
        float pv = acc[obi][r] + ((o < PROJ) ? pb[o] : 0.0f);
        sA[sidx(o, col)] = pv * maskc;
      }
    }
  }
  __syncthreads();

  // ---- spline: one thread per (q, column) ----
  const int q   = tid >> 7;            // 0..1
  const int col = tid & (TTILE - 1);
  const int tg  = t0 + col;
  const float maskv = xmask[(size_t)b * T_ + tg];

  float p[3 * NBINS - 1];
#pragma unroll
  for (int r = 0; r < 3 * NBINS - 1; ++r) p[r] = sA[sidx(q * (3 * NBINS - 1) + r, col)];

  const float scale = 0.0625f;  // 1/sqrt(256)
  const float c1 = 1.0f - MINWV * NBINS;

  // widths: softmax -> affine -> cumulative knots
  float uw[NBINS];
#pragma unroll
  for (int i = 0; i < NBINS; ++i) uw[i] = p[i] * scale;
  float mw = uw[0];
#pragma unroll
  for (int i = 1; i < NBINS; ++i) mw = fmaxf(mw, uw[i]);
  float ew[NBINS]; float sw = 0.0f;
#pragma unroll
  for (int i = 0; i < NBINS; ++i) { ew[i] = expf(uw[i] - mw); sw += ew[i]; }
  float invw = 1.0f / sw;
  float cwv[NBINS + 1];
  cwv[0] = -TAILV;
  {
    float run = 0.0f;
#pragma unroll
    for (int i = 0; i < NBINS; ++i) {
      run += MINWV + c1 * ew[i] * invw;
      cwv[i + 1] = 2.0f * TAILV * run - TAILV;
    }
  }
  cwv[NBINS] = TAILV;

  // heights
  float uh[NBINS];
#pragma unroll
  for (int i = 0; i < NBINS; ++i) uh[i] = p[NBINS + i] * scale;
  float mh2 = uh[0];
#pragma unroll
  for (int i = 1; i < NBINS; ++i) mh2 = fmaxf(mh2, uh[i]);
  float eh[NBINS]; float sh = 0.0f;
#pragma unroll
  for (int i = 0; i < NBINS; ++i) { eh[i] = expf(uh[i] - mh2); sh += eh[i]; }
  float invh = 1.0f / sh;
  float chv[NBINS + 1];
  chv[0] = -TAILV;
  {
    float run = 0.0f;
#pragma unroll
    for (int i = 0; i < NBINS; ++i) {
      run += MINWV + c1 * eh[i] * invh;
      chv[i + 1] = 2.0f * TAILV * run - TAILV;
    }
  }
  chv[NBINS] = TAILV;

  // derivatives (padded ends -> exactly 1.0)
  float dv[NBINS + 1];
  dv[0] = 1.0f; dv[NBINS] = 1.0f;
#pragma unroll
  for (int i = 0; i < NBINS - 1; ++i) dv[i + 1] = 0.001f + softplusf(p[2 * NBINS + i]);

  float x1v = x[((size_t)b * CIN + HALFC + q) * T_ + tg];
  float xin = fminf(fmaxf(x1v, -TAILV), TAILV);
  bool inside = (x1v >= -TAILV) && (x1v <= TAILV);

  int cnt = 0;
#pragma unroll
  for (int k = 0; k <= NBINS; ++k) cnt += (xin >= cwv[k]) ? 1 : 0;
  int bidx = cnt - 1;
  bidx = (bidx < 0) ? 0 : ((bidx > NBINS - 1) ? NBINS - 1 : bidx);

  float icw = 0.f, ibw = 1.f, ich = 0.f, ihb = 1.f, id0 = 1.f, id1 = 1.f;
#pragma unroll
  for (int i = 0; i < NBINS; ++i) {
    if (i == bidx) {
      icw = cwv[i]; ibw = cwv[i + 1] - cwv[i];
      ich = chv[i]; ihb = chv[i + 1] - chv[i];
      id0 = dv[i];  id1 = dv[i + 1];
    }
  }
  float idelta = ihb / ibw;
  float theta  = (xin - icw) / ibw;
  float omt    = 1.0f - theta;
  float tmt    = theta * omt;
  float num    = ihb * (idelta * theta * theta + id0 * tmt);
  float den    = idelta + (id0 + id1 - 2.0f * idelta) * tmt;
  float outv   = ich + num / den;
  float dnum   = idelta * idelta *
                 (id1 * theta * theta + 2.0f * idelta * tmt + id0 * omt * omt);
  float lad = logf(dnum) - 2.0f * logf(den);
  if (!inside) { outv = x1v; lad = 0.0f; }

  out[((size_t)b * CIN + q) * T_ + tg]          = x[((size_t)b * CIN + q) * T_ + tg] * maskv;
  out[((size_t)b * CIN + HALFC + q) * T_ + tg]  = outv * maskv;

  sRed[tid] = lad * maskv;
  __syncthreads();
  for (int s = 128; s > 0; s >>= 1) {
    if (tid < s) sRed[tid] += sRed[tid + s];
    __syncthreads();
  }
  if (tid == 0) part[blockIdx.x] = sRed[0];
}

// --------------------------------------------------------------------------
// Kernel 4: reduce per-tile logdet partials -> d_out tail
// --------------------------------------------------------------------------
__global__ void logdet_kernel(const float* __restrict__ part, float* __restrict__ ld) {
  __shared__ float s[NTILES];
  int b = blockIdx.x;
  s[threadIdx.x] = part[b * NTILES + threadIdx.x];
  __syncthreads();
  for (int st = NTILES / 2; st > 0; st >>= 1) {
    if ((int)threadIdx.x < st) s[threadIdx.x] += s[threadIdx.x + st];
    __syncthreads();
  }
  if (threadIdx.x == 0) ld[b] = s[0];
}

// --------------------------------------------------------------------------
extern "C" void kernel_launch(void* const* d_in, const int* in_sizes, int n_in,
                              void* d_out, int out_size, void* d_ws, size_t ws_size,
                              hipStream_t stream) {
  (void)in_sizes; (void)n_in; (void)out_size; (void)ws_size;
  const float* x      = (const float*)d_in[0];
  const float* xmask  = (const float*)d_in[1];
  const float* pre_w  = (const float*)d_in[2];
  const float* pre_b  = (const float*)d_in[3];
  const float* sep_w  = (const float*)d_in[4];
  const float* sep_b  = (const float*)d_in[5];
  const float* w1     = (const float*)d_in[6];
  const float* b1     = (const float*)d_in[7];
  const float* g1     = (const float*)d_in[8];
  const float* be1    = (const float*)d_in[9];
  const float* g2     = (const float*)d_in[10];
  const float* be2    = (const float*)d_in[11];
  const float* proj_w = (const float*)d_in[12];
  const float* proj_b = (const float*)d_in[13];
  float* out = (float*)d_out;

  const size_t HBUF = (size_t)B_ * FILT * T_;   // 33,554,432 floats
  float* hA   = (float*)d_ws;
  float* hB   = hA + HBUF;
  float* part = hB + HBUF;                      // B_*NTILES = 1024 floats

  const size_t total = HBUF;
  pre_kernel<<<(unsigned)((total + 255) / 256), 256, 0, stream>>>(x, pre_w, pre_b, hA);

  dim3 grid(B_ * NTILES);   // 1024 workgroups
  layer_kernel<<<grid, 256, 0, stream>>>(hA, hB, xmask, sep_w, sep_b, w1, b1,
                                         g1, be1, g2, be2, 0, 1);
  layer_kernel<<<grid, 256, 0, stream>>>(hB, hA, xmask, sep_w, sep_b, w1, b1,
                                         g1, be1, g2, be2, 1, 3);
  layer_kernel<<<grid, 256, 0, stream>>>(hA, hB, xmask, sep_w, sep_b, w1, b1,
                                         g1, be1, g2, be2, 2, 9);

  proj_spline_kernel<<<grid, 256, 0, stream>>>(hB, x, xmask, proj_w, proj_b, out, part);
  logdet_kernel<<<B_, NTILES, 0, stream>>>(part, out + (size_t)B_ * CIN * T_);
}